// SimpleGAT_28363964023689
// MI455X (gfx1250) — compile-verified
//
#include <hip/hip_runtime.h>

#define IN_CH 128
#define HEADS 4
#define HID 32
#define HC 128
#define NEG_SLOPE 0.2f

typedef __attribute__((ext_vector_type(2))) float v2f;
typedef __attribute__((ext_vector_type(8))) float v8f;

// ---------- monotonic float<->uint key for atomicMax over signed floats ----
__device__ __forceinline__ unsigned f2key(float f) {
    unsigned b = __float_as_uint(f);
    return (b & 0x80000000u) ? ~b : (b | 0x80000000u);
}
__device__ __forceinline__ float key2f(unsigned k) {
    return __uint_as_float((k & 0x80000000u) ? (k & 0x7fffffffu) : ~k);
}
__device__ __forceinline__ float lrelu(float x) {
    return x > 0.0f ? x : NEG_SLOPE * x;
}

// ---------------------------------------------------------------------------
// GEMM: Out[N,128] = In[N,128] @ Wm[128,128]  (+bias, +relu optional)
// One 256-thread block = 8 waves; each wave owns a 16-row slab x 128 cols
// (8 accumulator tiles), K stepped by 4 with V_WMMA_F32_16X16X4_F32.
//
// Fragment layouts (ISA 7.12.2, 32-bit):
//   A 16x4 : lane = (k>>1)*16 + m, vgpr = k&1
//   B 4x16 : lane = (k>>1)*16 + n, vgpr = k&1
//   C/D    : vgpr i : lanes 0-15 -> M=i, lanes 16-31 -> M=i+8; N = lane&15
//
// W is staged in LDS pair-interleaved + bank-swizzled:
//   dword_off(k, n) = (k>>1)*256 + 2*(n ^ (((k>>1)&1)*16)) + (k&1)
// so each B fragment is ONE aligned ds_load_b64 (no register fixups), and
// lanes 0-15 (even pair-row) / 16-31 (odd pair-row) touch disjoint bank
// halves -> conflict-free. Fragment base pointers are hoisted; the k-step
// offset (kb*512 B, max 63488) folds into the 16-bit DS immediate.
// ---------------------------------------------------------------------------
__global__ __launch_bounds__(256)
void gat_gemm128_wmma(const float* __restrict__ In, const float* __restrict__ Wm,
                      const float* __restrict__ bias, float* __restrict__ Out,
                      int N, int fuse_relu)
{
    __shared__ float sW[128 * 128];           // 64 KB of the 320 KB WGP LDS
    const int tid = threadIdx.x;

    // cooperative swizzled stage: thread handles (pair-row r, col n)
    for (int idx = tid; idx < 64 * 128; idx += 256) {
        const int r = idx >> 7;               // k pair-row 0..63
        const int n = idx & 127;              // column
        v2f w;
        w.x = Wm[(2 * r) * 128 + n];
        w.y = Wm[(2 * r + 1) * 128 + n];
        *(v2f*)(sW + r * 256 + 2 * (n ^ ((r & 1) * 16))) = w;
    }
    __syncthreads();

    const int wave = tid >> 5;
    const int lane = tid & 31;
    const int m    = lane & 15;               // row-in-tile / col-in-tile
    const int kg   = lane >> 4;               // k pair-group (0 or 1)
    const long row0 = ((long)blockIdx.x * 8 + wave) * 16;
    if (row0 >= N) return;                    // barrier already passed

    // per-lane B fragment base pointers (pair-row parity inside a k-step == kg)
    const float* pB[8];
#pragma unroll
    for (int j = 0; j < 8; ++j)
        pB[j] = sW + kg * 256 + 2 * ((j * 16 + m) ^ (kg * 16));

    v8f acc[8];
#pragma unroll
    for (int j = 0; j < 8; ++j)
#pragma unroll
        for (int i = 0; i < 8; ++i) acc[j][i] = 0.0f;

    long rA = row0 + m; if (rA >= N) rA = N - 1;   // clamped duplicate row
    const float* Arow = In + rA * 128;

    for (int kb = 0; kb < 128; kb += 4) {
        v2f a = *(const v2f*)(Arow + kb + 2 * kg); // x[m][ka], x[m][ka+1]
#pragma unroll
        for (int j = 0; j < 8; ++j) {
            v2f b = *(const v2f*)(pB[j] + kb * 128);   // one ds_load_b64
            acc[j] = __builtin_amdgcn_wmma_f32_16x16x4_f32(
                false, a, false, b, (short)0, acc[j], false, false);
        }
    }

    // store D: vgpr i -> row row0 + i + 8*kg, col j*16 + m
#pragma unroll
    for (int j = 0; j < 8; ++j) {
        const int c = j * 16 + m;
        const float bv = bias ? bias[c] : 0.0f;
#pragma unroll
        for (int i = 0; i < 8; ++i) {
            const long r = row0 + i + 8 * kg;
            if (r < N) {
                float v = acc[j][i] + bv;
                if (fuse_relu) v = fmaxf(v, 0.0f);
                Out[r * 128 + c] = v;
            }
        }
    }
}

// --------------------- per-node attention logit components -----------------
__global__ __launch_bounds__(256)
void gat_att_logits(const float* __restrict__ h, const float* __restrict__ att_src,
                    const float* __restrict__ att_dst,
                    float* __restrict__ a_src, float* __restrict__ a_dst, int N)
{
    const int n = blockIdx.x * blockDim.x + threadIdx.x;
    if (n >= N) return;
    const float4* hv = (const float4*)(h + (long)n * 128);
    const float4* as = (const float4*)att_src;
    const float4* ad = (const float4*)att_dst;
#pragma unroll
    for (int hd = 0; hd < HEADS; ++hd) {
        float ss = 0.0f, sd = 0.0f;
#pragma unroll
        for (int q = 0; q < 8; ++q) {
            float4 x = hv[hd * 8 + q];
            float4 u = as[hd * 8 + q];
            float4 v = ad[hd * 8 + q];
            ss += x.x * u.x + x.y * u.y + x.z * u.z + x.w * u.w;
            sd += x.x * v.x + x.y * v.y + x.z * v.z + x.w * v.w;
        }
        a_src[n * 4 + hd] = ss;
        a_dst[n * 4 + hd] = sd;
    }
}

// --------------------- edge pass A: segment max (atomicMax on keys) --------
__global__ __launch_bounds__(256)
void gat_edge_max(const int* __restrict__ src, const int* __restrict__ dst,
                  const float* __restrict__ a_src, const float* __restrict__ a_dst,
                  unsigned* __restrict__ mkey, int E)
{
    const int e = blockIdx.x * blockDim.x + threadIdx.x;
    if (e >= E) return;
    const int s = src[e], d = dst[e];
    float4 as = *(const float4*)(a_src + (long)s * 4);
    float4 ad = *(const float4*)(a_dst + (long)d * 4);
    unsigned* mp = mkey + (long)d * 4;
    atomicMax(mp + 0, f2key(lrelu(as.x + ad.x)));
    atomicMax(mp + 1, f2key(lrelu(as.y + ad.y)));
    atomicMax(mp + 2, f2key(lrelu(as.z + ad.z)));
    atomicMax(mp + 3, f2key(lrelu(as.w + ad.w)));
}

// --------------------- edge pass B: exp-weighted scatter-add ----------------
// One wave (32 lanes) per edge: lane g handles channels [4g,4g+4), head g>>3.
// acc[dst] += exp(e - m[dst]) * h[src]; s[dst] += exp(...) (once per head).
__global__ __launch_bounds__(256)
void gat_edge_scatter(const int* __restrict__ src, const int* __restrict__ dst,
                      const float* __restrict__ a_src, const float* __restrict__ a_dst,
                      const unsigned* __restrict__ mkey, const float* __restrict__ h,
                      float* __restrict__ ssum, float* __restrict__ acc, int E)
{
    const long t = (long)blockIdx.x * blockDim.x + threadIdx.x;
    const int e = (int)(t >> 5);
    const int g = (int)(t & 31);
    if (e >= E) return;
    const int s  = src[e], d = dst[e];
    const int hd = g >> 3;

    float l  = lrelu(a_src[(long)s * 4 + hd] + a_dst[(long)d * 4 + hd]);
    float mx = key2f(mkey[(long)d * 4 + hd]);
    float ex = __expf(l - mx);

    if ((g & 7) == 0) unsafeAtomicAdd(&ssum[(long)d * 4 + hd], ex);

    float4 hv = *(const float4*)(h + (long)s * 128 + g * 4);
    float* ap = acc + (long)d * 128 + g * 4;
    unsafeAtomicAdd(ap + 0, ex * hv.x);
    unsafeAtomicAdd(ap + 1, ex * hv.y);
    unsafeAtomicAdd(ap + 2, ex * hv.z);
    unsafeAtomicAdd(ap + 3, ex * hv.w);
}

// --------------------- deferred softmax normalization ----------------------
__global__ __launch_bounds__(256)
void gat_normalize(float* __restrict__ acc, const float* __restrict__ ssum, long total)
{
    const long i = (long)blockIdx.x * blockDim.x + threadIdx.x;
    if (i >= total) return;
    const long n = i >> 7;
    const int hd = (int)((i & 127) >> 5);
    acc[i] = acc[i] / (ssum[n * 4 + hd] + 1e-16f);
}

// ---------------------------------------------------------------------------
extern "C" void kernel_launch(void* const* d_in, const int* in_sizes, int n_in,
                              void* d_out, int out_size, void* d_ws, size_t ws_size,
                              hipStream_t stream)
{
    const float* x        = (const float*)d_in[0];
    const int*   eidx     = (const int*)  d_in[1];
    const float* W        = (const float*)d_in[2];
    const float* att_src  = (const float*)d_in[3];
    const float* att_dst  = (const float*)d_in[4];
    const float* lin_w    = (const float*)d_in[5];
    const float* lin_b    = (const float*)d_in[6];

    const int N = in_sizes[0] / IN_CH;   // 100000
    const int E = in_sizes[1] / 2;       // 1600000
    const int* src = eidx;
    const int* dst = eidx + E;

    // workspace carve-up (all 16B-aligned given N*128*4 | 16 and N*4*4 | 16)
    float*    h     = (float*)d_ws;                       // N*128
    float*    a_src = h     + (size_t)N * 128;            // N*4
    float*    a_dst = a_src + (size_t)N * 4;              // N*4
    unsigned* mkey  = (unsigned*)(a_dst + (size_t)N * 4); // N*4
    float*    ssum  = (float*)(mkey + (size_t)N * 4);     // N*4
    float*    acc   = ssum  + (size_t)N * 4;              // N*128

    // zero mkey (key 0 < any real value's key), ssum, acc in one memset
    hipMemsetAsync(mkey, 0, (size_t)N * (4 + 4 + 128) * sizeof(float), stream);

    const int gemm_blocks = (N + 127) / 128;

    // 1) h = x @ W  (fp32 WMMA)
    gat_gemm128_wmma<<<gemm_blocks, 256, 0, stream>>>(x, W, nullptr, h, N, 0);
    // 2) per-node logit components
    gat_att_logits<<<(N + 255) / 256, 256, 0, stream>>>(h, att_src, att_dst, a_src, a_dst, N);
    // 3) segment max over destination neighborhoods
    gat_edge_max<<<(E + 255) / 256, 256, 0, stream>>>(src, dst, a_src, a_dst, mkey, E);
    // 4) exp-weighted scatter-sum + softmax denominator
    const long tB = (long)E * 32;
    gat_edge_scatter<<<(int)((tB + 255) / 256), 256, 0, stream>>>(src, dst, a_src, a_dst,
                                                                  mkey, h, ssum, acc, E);
    // 5) normalize: acc /= (s + eps)
    const long tot = (long)N * 128;
    gat_normalize<<<(int)((tot + 255) / 256), 256, 0, stream>>>(acc, ssum, tot);
    // 6) out = relu(acc @ lin_w + b)  (fp32 WMMA, fused bias+relu)
    gat_gemm128_wmma<<<gemm_blocks, 256, 0, stream>>>(acc, lin_w, lin_b, (float*)d_out, N, 1);
}